// ExpandOperator_5531917878017
// MI455X (gfx1250) — compile-verified
//
#include <hip/hip_runtime.h>

// ---------------- problem constants (from reference) ----------------
#define B_SZ      2
#define S_LEN     2048
#define D_IN      768
#define MAX_SPAN  64
#define SPAN_DIM  32
#define N_TOT     (MAX_SPAN * SPAN_DIM)   // 2048, GEMM N
#define M_TOT     (B_SZ * S_LEN)          // 4096, GEMM M

// ---------------- tile config ----------------
#define BM 64
#define BN 128
#define BK 32
#define NCHUNK (D_IN / BK)                // 24

typedef __attribute__((ext_vector_type(2))) float v2f;
typedef __attribute__((ext_vector_type(4))) float v4f;
typedef __attribute__((ext_vector_type(8))) float v8f;

// ---------------------------------------------------------------
// Kernel 1: zero-fill the 1.07 GB output with non-temporal b128
// stores (streaming write-once data must not thrash the 192 MB L2).
// Each thread writes 4 coalesced float4's.
// ---------------------------------------------------------------
__global__ __launch_bounds__(256)
void zero_out_kernel(v4f* __restrict__ out, long n4) {
    long base = (long)blockIdx.x * (blockDim.x * 4) + threadIdx.x;
    v4f z = {};
    #pragma unroll
    for (int j = 0; j < 4; ++j) {
        long i = base + (long)j * 256;
        if (i < n4) __builtin_nontemporal_store(z, &out[i]);
    }
}

// ---------------------------------------------------------------
// Kernel 2: fp32 WMMA GEMM (M=4096, N=2048, K=768) + banded scatter,
// LDS double-buffered (one barrier per K-chunk), NT epilogue stores.
//   pred[m][n] = sum_k X[m][k] * W[n][k] + bias[n]
//   out[b, t, (t + n/32) % S, n%32] = pred[m][n],  m = b*S + t
// ---------------------------------------------------------------
__global__ __launch_bounds__(256)
void band_gemm_kernel(const float* __restrict__ X,     // (4096, 768) row-major
                      const float* __restrict__ W,     // (2048, 768) row-major
                      const float* __restrict__ bias,  // (2048)
                      float* __restrict__ out)         // (B, S, S, 32)
{
    __shared__ float Xs[2][BM][BK];   // 2 x  8 KB
    __shared__ float Ws[2][BN][BK];   // 2 x 16 KB   (Ws[.][n][k] == B[k][n])

    const int tid    = threadIdx.x;
    const int lane   = tid & 31;
    const int wave   = tid >> 5;        // 0..7
    const int wave_m = wave & 3;        // 4 waves along M
    const int wave_n = wave >> 2;       // 2 waves along N
    const int half   = lane >> 4;       // lane-half selects K pair
    const int l16    = lane & 15;

    const int m0 = blockIdx.x * BM;     // gridDim.x = 64
    const int n0 = blockIdx.y * BN;     // gridDim.y = 16

    // per-thread staging slots (same for every chunk)
    const int xrow = tid >> 3;                 // tid..tid+256: rows 0..63
    const int xc4  = (tid & 7) * 4;
    v8f acc[4] = {};                           // 16x64 C strip per wave
    v4f xr[2], wr[4];

    // ---- preload chunk 0 into registers, then LDS buf 0 ----
    #pragma unroll
    for (int r = 0; r < 2; ++r) {
        int idx = tid + r * 256, row = idx >> 3, c4 = (idx & 7) * 4;
        xr[r] = *(const v4f*)(X + (size_t)(m0 + row) * D_IN + c4);
    }
    #pragma unroll
    for (int r = 0; r < 4; ++r) {
        int idx = tid + r * 256, row = idx >> 3, c4 = (idx & 7) * 4;
        wr[r] = *(const v4f*)(W + (size_t)(n0 + row) * D_IN + c4);
    }
    #pragma unroll
    for (int r = 0; r < 2; ++r) {
        int idx = tid + r * 256, row = idx >> 3, c4 = (idx & 7) * 4;
        *(v4f*)&Xs[0][row][c4] = xr[r];
    }
    #pragma unroll
    for (int r = 0; r < 4; ++r) {
        int idx = tid + r * 256, row = idx >> 3, c4 = (idx & 7) * 4;
        *(v4f*)&Ws[0][row][c4] = wr[r];
    }
    __syncthreads();

    for (int c = 0; c < NCHUNK; ++c) {
        const int cur = c & 1;
        const bool has_next = (c + 1) < NCHUNK;

        // ---- issue next chunk's global loads (overlap with WMMA) ----
        if (has_next) {
            const int kc = (c + 1) * BK;
            #pragma unroll
            for (int r = 0; r < 2; ++r) {
                int idx = tid + r * 256, row = idx >> 3, c4 = (idx & 7) * 4;
                xr[r] = *(const v4f*)(X + (size_t)(m0 + row) * D_IN + kc + c4);
            }
            #pragma unroll
            for (int r = 0; r < 4; ++r) {
                int idx = tid + r * 256, row = idx >> 3, c4 = (idx & 7) * 4;
                wr[r] = *(const v4f*)(W + (size_t)(n0 + row) * D_IN + kc + c4);
            }
        }

        // ---- 8 K-steps of 4, 4 N-fragments each: 32 WMMA / chunk ----
        #pragma unroll
        for (int k = 0; k < BK; k += 4) {
            // A frag 16x4 f32: lane<16 -> (M=l16, K=k..k+1), lane>=16 -> K=k+2..k+3
            v2f a = *(const v2f*)&Xs[cur][wave_m * 16 + l16][k + half * 2];
            #pragma unroll
            for (int f = 0; f < 4; ++f) {
                // B frag 4x16 f32: same K-pair split across lane halves
                v2f b = *(const v2f*)&Ws[cur][wave_n * 64 + f * 16 + l16][k + half * 2];
                acc[f] = __builtin_amdgcn_wmma_f32_16x16x4_f32(
                    /*neg_a=*/false, a, /*neg_b=*/false, b,
                    /*c_mod=*/(short)0, acc[f],
                    /*reuse_a=*/false, /*reuse_b=*/false);
            }
        }

        // ---- commit next chunk to the alternate LDS buffer ----
        if (has_next) {
            const int nxt = cur ^ 1;
            #pragma unroll
            for (int r = 0; r < 2; ++r) {
                int idx = tid + r * 256, row = idx >> 3, c4 = (idx & 7) * 4;
                *(v4f*)&Xs[nxt][row][c4] = xr[r];
            }
            #pragma unroll
            for (int r = 0; r < 4; ++r) {
                int idx = tid + r * 256, row = idx >> 3, c4 = (idx & 7) * 4;
                *(v4f*)&Ws[nxt][row][c4] = wr[r];
            }
        }
        __syncthreads();   // single barrier per chunk
    }
    (void)xrow; (void)xc4;

    // ---- epilogue: bias + banded scatter, non-temporal stores ----
    // C layout: VGPR r, lanes 0-15 -> (M=r, N=l16); lanes 16-31 -> (M=r+8, N=l16)
    #pragma unroll
    for (int f = 0; f < 4; ++f) {
        const int n_glob = n0 + wave_n * 64 + f * 16 + l16;
        const int span   = n_glob >> 5;          // n / 32
        const int d      = n_glob & 31;          // n % 32
        const float bv   = bias[n_glob];
        #pragma unroll
        for (int r = 0; r < 8; ++r) {
            const int m_glob = m0 + wave_m * 16 + r + half * 8;
            const int bb = m_glob >> 11;         // / S_LEN
            const int t  = m_glob & (S_LEN - 1);
            const int j  = (t + span) & (S_LEN - 1);
            const size_t oi = ((((size_t)bb * S_LEN + t) * S_LEN) + j) * SPAN_DIM + d;
            __builtin_nontemporal_store(acc[f][r] + bv, &out[oi]);
        }
    }
}

// ---------------------------------------------------------------
extern "C" void kernel_launch(void* const* d_in, const int* in_sizes, int n_in,
                              void* d_out, int out_size, void* d_ws, size_t ws_size,
                              hipStream_t stream) {
    const float* x  = (const float*)d_in[0];   // (2, 2048, 768)
    const float* W  = (const float*)d_in[1];   // (2048, 768)
    const float* bv = (const float*)d_in[2];   // (2048,)
    float* out = (float*)d_out;                // (2, 2048, 2048, 32) fp32

    // 1) zero-fill full output (harness poisons it with 0xAA)
    long n4 = (long)out_size / 4;              // 67,108,864 float4
    long zb = (n4 + (256 * 4 - 1)) / (256 * 4);
    zero_out_kernel<<<dim3((unsigned)zb), 256, 0, stream>>>((v4f*)out, n4);

    // 2) WMMA GEMM + band scatter
    dim3 grid(M_TOT / BM, N_TOT / BN);         // (64, 16)
    band_gemm_kernel<<<grid, 256, 0, stream>>>(x, W, bv, out);
}